// CurvatureLoss_67920612819270
// MI455X (gfx1250) — compile-verified
//
#include <hip/hip_runtime.h>

// CurvatureLoss for CDNA5 (gfx1250, wave32).
//
// Core: brute-force KNN via V_WMMA_F32_16X16X4_F32.
//   A (16x4, per ISA layout: lanes 0-15 supply K=0,1; lanes 16-31 supply K=2,3
//      for row M = lane%16):  row m = ( rx, ry, rz, |r|^2 )
//   B (4x16, symmetric):      col n = ( -2qx, -2qy, -2qz, 1 )
//   => D[m][n] = |r_m|^2 - 2 r_m . q_n ; add per-lane |q_n|^2 afterwards.
//
// D layout (ISA 7.12.2): VGPR v: lanes 0-15 -> M=v, lanes 16-31 -> M=v+8, N=lane%16.
// So lane li and lane li+16 both own query li; each tracks top-K over its 8
// candidates/tile, lists merged across the pair with __shfl_xor(...,16).
//
// Tile loop is batched 4x: four independent global_load_b96 in flight per
// iteration (staged s_wait_loadcnt instead of a serial load->wait->use chain),
// plus a global_prefetch two batches ahead.

typedef __attribute__((ext_vector_type(2))) float v2f;
typedef __attribute__((ext_vector_type(8))) float v8f;

#define RADIUS2 2.5f
#define TILE_BATCH 4

template<int K>
__device__ __forceinline__ void topk_insert(float (&kd)[K], int (&ki)[K], float d, int ci) {
  // kd sorted ascending; replace tail and bubble down. Fully unrolled -> registers.
  if (d < kd[K - 1]) {
    kd[K - 1] = d; ki[K - 1] = ci;
#pragma unroll
    for (int j = K - 1; j > 0; --j) {
      if (kd[j] < kd[j - 1]) {
        float td = kd[j]; kd[j] = kd[j - 1]; kd[j - 1] = td;
        int   ti = ki[j]; ki[j] = ki[j - 1]; ki[j - 1] = ti;
      }
    }
  }
}

// Wave-cooperative KNN: 16 queries per wave. li = lane&15, half = lane>>4.
// On exit every lane holds the merged global top-K (ascending) for query li.
template<int K>
__device__ __forceinline__ void knn16_wmma(const float* __restrict__ ref, int N,
                                           int li, int half,
                                           float qx, float qy, float qz,
                                           float (&kd)[K], int (&ki)[K]) {
  float qn = qx * qx + qy * qy + qz * qz;
  v2f bq;                                  // loop-invariant B operand
  bq.x = half ? (-2.0f * qz) : (-2.0f * qx);
  bq.y = half ? 1.0f         : (-2.0f * qy);

#pragma unroll
  for (int j = 0; j < K; ++j) { kd[j] = 3.4e38f; ki[j] = 0; }

  // N % (16*TILE_BATCH) == 0 for this problem (N = 8192).
  for (int tb = 0; tb < N; tb += 16 * TILE_BATCH) {
    // Prefetch two batches ahead (global_prefetch_b8; wrap to keep addr valid).
    int pf = tb + 32 * TILE_BATCH;
    pf = (pf < N) ? pf : 0;
    __builtin_prefetch(ref + (size_t)(pf + li) * 3, 0, 1);

    // Issue all 4 tile loads before consuming any (keeps 4 loads in flight).
    float rxy[TILE_BATCH][3];
#pragma unroll
    for (int i = 0; i < TILE_BATCH; ++i) {
      const float* rp = ref + (size_t)(tb + 16 * i + li) * 3;
      rxy[i][0] = rp[0]; rxy[i][1] = rp[1]; rxy[i][2] = rp[2];
    }

#pragma unroll
    for (int i = 0; i < TILE_BATCH; ++i) {
      float rx = rxy[i][0], ry = rxy[i][1], rz = rxy[i][2];
      float rn = rx * rx + ry * ry + rz * rz;
      v2f a;
      a.x = half ? rz : rx;
      a.y = half ? rn : ry;
      v8f c = {};
      c = __builtin_amdgcn_wmma_f32_16x16x4_f32(false, a, false, bq, (short)0, c,
                                                false, false);
      int base = tb + 16 * i + 8 * half;
#pragma unroll
      for (int v = 0; v < 8; ++v) {
        float d  = c[v] + qn;               // full squared distance
        int   ci = base + v;                // ref index for this VGPR/half
        topk_insert<K>(kd, ki, d, ci);
      }
    }
  }

  // Merge the two half-subset lists across the lane pair (li, li+16). Snapshot
  // partner first (shuffles happen before either side mutates its list).
  float pd[K]; int pi[K];
#pragma unroll
  for (int j = 0; j < K; ++j) {
    pd[j] = __shfl_xor(kd[j], 16, 32);
    pi[j] = __shfl_xor(ki[j], 16, 32);
  }
#pragma unroll
  for (int j = 0; j < K; ++j) topk_insert<K>(kd, ki, pd[j], pi[j]);
}

__device__ __forceinline__ void wave_ids(int N, int& li, int& half, int& b, int& q) {
  int tid  = blockIdx.x * blockDim.x + threadIdx.x;
  int wave = tid >> 5;
  int lane = tid & 31;
  li = lane & 15; half = lane >> 4;
  int qglob = wave * 16 + li;   // N % 16 == 0 -> waves never straddle batches
  b = qglob / N;
  q = qglob - b * N;
}

// curv2[n] = (sum_k pc2[idx_k(masked)] - 10*pc2[n]) / 9, idx from self-KNN k=10
__global__ void curv_self_kernel(const float* __restrict__ pts,
                                 float* __restrict__ curv, int N) {
  int li, half, b, q; wave_ids(N, li, half, b, q);
  const float* ref = pts + (size_t)b * N * 3;
  const float* qp  = ref + (size_t)q * 3;
  float qx = qp[0], qy = qp[1], qz = qp[2];
  float kd[10]; int ki[10];
  knn16_wmma<10>(ref, N, li, half, qx, qy, qz, kd, ki);
  if (half == 0) {
    float sx = 0.f, sy = 0.f, sz = 0.f;
    int i0 = ki[0];
#pragma unroll
    for (int j = 0; j < 10; ++j) {
      int id = (kd[j] > RADIUS2) ? i0 : ki[j];
      const float* p = ref + (size_t)id * 3;
      sx += p[0]; sy += p[1]; sz += p[2];
    }
    float* o = curv + ((size_t)b * N + q) * 3;
    o[0] = (sx - 10.0f * qx) * (1.0f / 9.0f);
    o[1] = (sy - 10.0f * qy) * (1.0f / 9.0f);
    o[2] = (sz - 10.0f * qz) * (1.0f / 9.0f);
  }
}

// curv1[n] = (sum_k warped[idx_k(masked)] - 10*warped[n]) / 9, idx from self-KNN on pc1
__global__ void curv_warp_kernel(const float* __restrict__ pc1,
                                 const float* __restrict__ flow,
                                 float* __restrict__ curv, int N) {
  int li, half, b, q; wave_ids(N, li, half, b, q);
  const float* ref = pc1  + (size_t)b * N * 3;
  const float* fl  = flow + (size_t)b * N * 3;
  const float* qp  = ref + (size_t)q * 3;
  float qx = qp[0], qy = qp[1], qz = qp[2];
  float kd[10]; int ki[10];
  knn16_wmma<10>(ref, N, li, half, qx, qy, qz, kd, ki);
  if (half == 0) {
    const float* fq = fl + (size_t)q * 3;
    float wx = qx + fq[0], wy = qy + fq[1], wz = qz + fq[2];
    float sx = 0.f, sy = 0.f, sz = 0.f;
    int i0 = ki[0];
#pragma unroll
    for (int j = 0; j < 10; ++j) {
      int id = (kd[j] > RADIUS2) ? i0 : ki[j];
      const float* p = ref + (size_t)id * 3;
      const float* f = fl  + (size_t)id * 3;
      sx += p[0] + f[0]; sy += p[1] + f[1]; sz += p[2] + f[2];
    }
    float* o = curv + ((size_t)b * N + q) * 3;
    o[0] = (sx - 10.0f * wx) * (1.0f / 9.0f);
    o[1] = (sy - 10.0f * wy) * (1.0f / 9.0f);
    o[2] = (sz - 10.0f * wz) * (1.0f / 9.0f);
  }
}

// KNN(warped -> pc2, k=5); inverse-distance interp of curv2; squared error vs
// curv1; deterministic per-block reduction -> partials[blockIdx].
__global__ void interp_loss_kernel(const float* __restrict__ pc1,
                                   const float* __restrict__ flow,
                                   const float* __restrict__ pc2,
                                   const float* __restrict__ curv2,
                                   const float* __restrict__ curv1,
                                   float* __restrict__ partials, int N) {
  __shared__ float sdata[256];
  int li, half, b, q; wave_ids(N, li, half, b, q);
  const float* ref = pc2   + (size_t)b * N * 3;
  const float* p1  = pc1   + (size_t)b * N * 3;
  const float* fl  = flow  + (size_t)b * N * 3;
  const float* c2  = curv2 + (size_t)b * N * 3;
  const float* c1  = curv1 + (size_t)b * N * 3;

  const float* qp = p1 + (size_t)q * 3;
  const float* fq = fl + (size_t)q * 3;
  float qx = qp[0] + fq[0], qy = qp[1] + fq[1], qz = qp[2] + fq[2];  // warped query

  float kd[5]; int ki[5];
  knn16_wmma<5>(ref, N, li, half, qx, qy, qz, kd, ki);

  float sq = 0.0f;
  if (half == 0) {
    float w[5]; float wsum = 0.0f;
#pragma unroll
    for (int j = 0; j < 5; ++j) { w[j] = 1.0f / (kd[j] + 1e-8f); wsum += w[j]; }
    float inv = 1.0f / wsum;
    int i0 = ki[0];
    float ix = 0.f, iy = 0.f, iz = 0.f;
#pragma unroll
    for (int j = 0; j < 5; ++j) {
      int id = (kd[j] > RADIUS2) ? i0 : ki[j];   // weights use UNMASKED dists
      const float* c = c2 + (size_t)id * 3;
      float ww = w[j] * inv;
      ix += ww * c[0]; iy += ww * c[1]; iz += ww * c[2];
    }
    const float* m = c1 + (size_t)q * 3;
    float dx = ix - m[0], dy = iy - m[1], dz = iz - m[2];
    sq = dx * dx + dy * dy + dz * dz;
  }
  sdata[threadIdx.x] = sq;
  __syncthreads();
#pragma unroll
  for (int off = 128; off > 0; off >>= 1) {
    if ((int)threadIdx.x < off) sdata[threadIdx.x] += sdata[threadIdx.x + off];
    __syncthreads();
  }
  if (threadIdx.x == 0) partials[blockIdx.x] = sdata[0];
}

__global__ void finalize_kernel(const float* __restrict__ partials,
                                float* __restrict__ out, int nPart, float scale) {
  __shared__ float s[256];
  float v = 0.0f;
  for (int i = threadIdx.x; i < nPart; i += 256) v += partials[i];
  s[threadIdx.x] = v;
  __syncthreads();
#pragma unroll
  for (int off = 128; off > 0; off >>= 1) {
    if ((int)threadIdx.x < off) s[threadIdx.x] += s[threadIdx.x + off];
    __syncthreads();
  }
  if (threadIdx.x == 0) out[0] = s[0] * scale;
}

extern "C" void kernel_launch(void* const* d_in, const int* in_sizes, int n_in,
                              void* d_out, int out_size, void* d_ws, size_t ws_size,
                              hipStream_t stream) {
  const float* pc1  = (const float*)d_in[0];   // pc_source  [B,N,3] f32
  const float* pc2  = (const float*)d_in[1];   // pc_target  [B,N,3] f32
  const float* flow = (const float*)d_in[2];   // pred_flow  [B,N,3] f32

  const int B = 2;
  const int N = in_sizes[0] / (B * 3);         // 8192

  float* ws       = (float*)d_ws;
  size_t BN3      = (size_t)B * N * 3;
  float* curv2    = ws;                        // [B,N,3]
  float* curv1    = ws + BN3;                  // [B,N,3]
  float* partials = ws + 2 * BN3;              // [blocks]

  const int threads = 256;                     // 8 wave32s per block
  const int blocks  = (B * N) / 128;           // 16 queries per wave * 8 waves

  curv_self_kernel<<<blocks, threads, 0, stream>>>(pc2, curv2, N);
  curv_warp_kernel<<<blocks, threads, 0, stream>>>(pc1, flow, curv1, N);
  interp_loss_kernel<<<blocks, threads, 0, stream>>>(pc1, flow, pc2, curv2, curv1,
                                                     partials, N);
  finalize_kernel<<<1, 256, 0, stream>>>(partials, (float*)d_out, blocks,
                                         1.0f / (float)B);
}